// Query_2327872274828
// MI455X (gfx1250) — compile-verified
//
#include <hip/hip_runtime.h>
#include <float.h>

typedef __attribute__((ext_vector_type(2))) float v2f;
typedef __attribute__((ext_vector_type(8))) float v8f;

#define BLOCK   256
#define WAVES   8          // BLOCK / 32
#define NSPLIT  4          // waves co-scanning the same query group (N-split)
#define QGROUPS 2          // query groups of 16 per block  -> 32 queries/block
#define QPB     (QGROUPS * 16)
#define CHUNK   1024       // coords staged in LDS per iteration (16 KB)
// tiles per chunk = CHUNK/16 = 64 ; per wave = 64/NSPLIT = 16

// dist(m,n) = ||c_n||^2 - 2 p_m . c_n  as a K=4 GEMM:
//   A[m] = [-2px, -2py, -2pz, 1],  B[:,n] = [cx, cy, cz, ||c||^2]
__global__ __launch_bounds__(BLOCK) void nnq_wmma_kernel(
    const float* __restrict__ coords,   // [N,3]
    const float* __restrict__ feature,  // [N,C]
    const float* __restrict__ points,   // [M,3]
    float* __restrict__ out,            // [M,C]
    int N, int M, int C)
{
    // Two float2 entries per coord: [2n]={cx,cy} (K=0,1), [2n+1]={cz,norm} (K=2,3)
    __shared__ float2 ldsB[CHUNK * 2];
    __shared__ float  pval[QPB][NSPLIT];
    __shared__ int    pidx[QPB][NSPLIT];
    __shared__ int    fidx[QPB];

    const int tid   = threadIdx.x;
    const int lane  = tid & 31;
    const int wave  = tid >> 5;
    const int lh    = lane >> 4;    // lane half: 0 -> K=0,1 ; 1 -> K=2,3
    const int l15   = lane & 15;
    const int qg    = wave >> 2;    // query group 0..1
    const int split = wave & 3;     // N-split id  0..3

    const int qblock = blockIdx.x * QPB;
    const int qbase  = qblock + qg * 16;

    // ---- A operand (fixed for whole sweep) ----
    int mq = qbase + l15;
    if (mq >= M) mq = M - 1;        // clamp (dup row; output store is guarded)
    const float px = points[mq * 3 + 0];
    const float py = points[mq * 3 + 1];
    const float pz = points[mq * 3 + 2];
    v2f a;
    a.x = lh ? (-2.0f * pz) : (-2.0f * px);
    a.y = lh ? (1.0f)       : (-2.0f * py);

    float minv[8];
    int   mini[8];
#pragma unroll
    for (int i = 0; i < 8; ++i) { minv[i] = FLT_MAX; mini[i] = 0; }

    // fixed part of the LDS read index (float2 units): 2*t + 2*l15 + lh
    const int ldoff = 2 * (split * 16 + l15) + lh;

    for (int cb = 0; cb < N; cb += CHUNK) {
        __syncthreads();
        // ---- cooperative stage: coords -> LDS (+ precomputed ||c||^2) ----
        for (int n = tid; n < CHUNK; n += BLOCK) {
            const int g = cb + n;
            float cx = 0.0f, cy = 0.0f, cz = 0.0f, nrm = FLT_MAX;
            if (g < N) {
                cx  = coords[g * 3 + 0];
                cy  = coords[g * 3 + 1];
                cz  = coords[g * 3 + 2];
                nrm = cx * cx + cy * cy + cz * cz;
            }
            ldsB[n * 2 + 0] = make_float2(cx, cy);
            ldsB[n * 2 + 1] = make_float2(cz, nrm);
        }
        __syncthreads();

        // ---- this wave sweeps 16 interleaved tiles of the chunk, 4 at a time ----
        for (int j = 0; j < 16; j += 4) {
            float2 bb[4];
#pragma unroll
            for (int u = 0; u < 4; ++u) {
                // tile t = split*16 + (j+u)*64 ; index = 2*t + 2*l15 + lh
                bb[u] = ldsB[ldoff + (j + u) * 128];
            }
#pragma unroll
            for (int u = 0; u < 4; ++u) {
                v2f b; b.x = bb[u].x; b.y = bb[u].y;
                v8f c = {};
                v8f d = __builtin_amdgcn_wmma_f32_16x16x4_f32(
                    false, a, false, b, (short)0, c, false, false);
                const int cand = cb + split * 16 + (j + u) * 64 + l15;
#pragma unroll
                for (int i = 0; i < 8; ++i) {
                    const float dv  = d[i];
                    const bool  upd = dv < minv[i];
                    minv[i] = upd ? dv   : minv[i];
                    mini[i] = upd ? cand : mini[i];
                }
            }
        }
    }

    // ---- xor-butterfly argmin across each 16-lane group (branchless) ----
#pragma unroll
    for (int off = 1; off < 16; off <<= 1) {
#pragma unroll
        for (int i = 0; i < 8; ++i) {
            const float ov = __shfl_xor(minv[i], off, 32);
            const int   oi = __shfl_xor(mini[i], off, 32);
            const bool  tk = (ov < minv[i]) | ((ov == minv[i]) & (oi < mini[i]));
            minv[i] = tk ? ov : minv[i];
            mini[i] = tk ? oi : mini[i];
        }
    }

    // ---- publish per-split partials (one lane per 16-lane group) ----
    __syncthreads();
    if (l15 == 0) {
        const int rbase = qg * 16 + 8 * lh;
#pragma unroll
        for (int i = 0; i < 8; ++i) {
            pval[rbase + i][split] = minv[i];
            pidx[rbase + i][split] = mini[i];
        }
    }
    __syncthreads();

    // ---- merge the NSPLIT partials per row (lexicographic, branchless) ----
    if (tid < QPB) {
        float bv = pval[tid][0];
        int   bi = pidx[tid][0];
#pragma unroll
        for (int s = 1; s < NSPLIT; ++s) {
            const float ov = pval[tid][s];
            const int   oi = pidx[tid][s];
            const bool  tk = (ov < bv) | ((ov == bv) & (oi < bi));
            bv = tk ? ov : bv;
            bi = tk ? oi : bi;
        }
        fidx[tid] = bi;
    }
    __syncthreads();

    // ---- gather: each wave copies 4 rows, full 32 lanes per row (b128) ----
    const int nf4 = C >> 2;   // float4s per row (C=128 -> 32 -> 1 iter/lane)
#pragma unroll
    for (int r = 0; r < QPB / WAVES; ++r) {
        const int row = wave * (QPB / WAVES) + r;
        const int m   = qblock + row;
        if (m >= M) continue;
        const int idx = fidx[row];
        const float4* __restrict__ src =
            (const float4*)(feature + (size_t)idx * (size_t)C);
        float4* __restrict__ dst = (float4*)(out + (size_t)m * (size_t)C);
        for (int k = lane; k < nf4; k += 32) {
            dst[k] = src[k];   // global_load_b128 / global_store_b128
        }
    }
}

extern "C" void kernel_launch(void* const* d_in, const int* in_sizes, int n_in,
                              void* d_out, int out_size, void* d_ws, size_t ws_size,
                              hipStream_t stream) {
    const float* coords  = (const float*)d_in[0];  // [N,3]
    const float* feature = (const float*)d_in[1];  // [N,C]
    const float* points  = (const float*)d_in[2];  // [M,3]
    float* out           = (float*)d_out;          // [M,C]

    const int N = in_sizes[0] / 3;
    const int M = in_sizes[2] / 3;
    const int C = in_sizes[1] / N;

    const int grid = (M + QPB - 1) / QPB;          // 32 queries per block
    nnq_wmma_kernel<<<grid, BLOCK, 0, stream>>>(coords, feature, points, out,
                                                N, M, C);
}